// LM_Softmax_48644799594926
// MI455X (gfx1250) — compile-verified
//
#include <hip/hip_runtime.h>
#include <cstdint>

#define NROWS 32768
#define NCOLS 2048
#define S_CONST 1.05f
#define RPB 8     // rows per block (software-pipelined)
#define TPB 256   // 8 waves of 32

// --- CDNA5 async global->LDS copy: 32 bytes per lane (2 x B128) ----------
// ISA 15.18.3 op 98: the instruction offset is added to both the LDS and the global address
// Non-temporal hint: 256MiB stream > 192MB L2, zero reuse -> don't pollute L2.
__device__ __forceinline__ void async_copy32B(const float* g, float* l) {
  uint32_t lo = (uint32_t)(uintptr_t)l;   // low 32 bits of flat shared ptr == LDS offset
  uint64_t go = (uint64_t)(uintptr_t)g;
  asm volatile("global_load_async_to_lds_b128 %0, %1, off th:TH_LOAD_NT"
               :: "v"(lo), "v"(go) : "memory");
  asm volatile("global_load_async_to_lds_b128 %0, %1, off offset:16 th:TH_LOAD_NT"
               :: "v"(lo), "v"(go) : "memory");
}

__device__ __forceinline__ float waveSum(float x) {
#pragma unroll
  for (int off = 16; off >= 1; off >>= 1) x += __shfl_xor(x, off, 32);
  return x;
}

// ---- zero the class accumulators (ws is poisoned, not re-zeroed) --------
__global__ void zeroK(float* p, int n) {
  int i = blockIdx.x * blockDim.x + threadIdx.x;
  if (i < n) p[i] = 0.0f;
}

// ---- main streaming pass: one read of the 256MiB matrix -----------------
__global__ __launch_bounds__(TPB) void rowpass(
    const float* __restrict__ X, const int* __restrict__ lbl,
    float* __restrict__ g_sum, float* __restrict__ g_sq, float* __restrict__ g_cnt,
    float* __restrict__ Mrow, float* __restrict__ Erow, float* __restrict__ Lrow) {
  __shared__ __align__(16) float buf[2][NCOLS];   // 16 KB double buffer
  __shared__ float red[4][8];                     // per-wave partials
  const int t = threadIdx.x;
  const int wid = t >> 5, lid = t & 31;
  const int row0 = blockIdx.x * RPB;

  // prologue: stage row0
  async_copy32B(X + (size_t)row0 * NCOLS + t * 8, &buf[0][t * 8]);

  for (int i = 0; i < RPB; ++i) {
    const int row = row0 + i;
    const float* bcur = buf[i & 1];

    if (i + 1 < RPB) {  // prefetch next row, then wait for current (in-order done)
      async_copy32B(X + (size_t)(row + 1) * NCOLS + t * 8, &buf[(i + 1) & 1][t * 8]);
      asm volatile("s_wait_asynccnt 2" ::: "memory");
    } else {
      asm volatile("s_wait_asynccnt 0" ::: "memory");
    }

    // each thread consumes exactly the 32B it staged (wave-private: no barrier
    // needed between the wait and these LDS reads)
    float4 a = *(const float4*)&bcur[t * 8];
    float4 b = *(const float4*)&bcur[t * 8 + 4];
    float v[8] = {a.x, a.y, a.z, a.w, b.x, b.y, b.z, b.w};

    float s = 0.f, q = 0.f, mx = -3.402823466e38f;
#pragma unroll
    for (int k = 0; k < 8; ++k) { s += v[k]; q += v[k] * v[k]; mx = fmaxf(mx, v[k]); }
#pragma unroll
    for (int off = 16; off >= 1; off >>= 1) {
      s += __shfl_xor(s, off, 32);
      q += __shfl_xor(q, off, 32);
      mx = fmaxf(mx, __shfl_xor(mx, off, 32));
    }
    if (lid == 0) { red[0][wid] = s; red[1][wid] = q; red[2][wid] = mx; }
    __syncthreads();                                         // B1
    float rs = 0.f, rq = 0.f, rm = -3.402823466e38f;
#pragma unroll
    for (int j = 0; j < 8; ++j) {
      rs += red[0][j]; rq += red[1][j]; rm = fmaxf(rm, red[2][j]);
    }

    float e = 0.f;
#pragma unroll
    for (int k = 0; k < 8; ++k) e += __expf(v[k] - rm);
    e = waveSum(e);
    if (lid == 0) red[3][wid] = e;
    __syncthreads();                                         // B2
    float re = 0.f;
#pragma unroll
    for (int j = 0; j < 8; ++j) re += red[3][j];

    if (t == 0) {
      int cl = lbl[row];
      float lc = bcur[cl];            // target logit, row still resident in LDS
      atomicAdd(&g_sum[cl], rs);
      atomicAdd(&g_sq[cl],  rq);
      atomicAdd(&g_cnt[cl], 1.0f);
      Mrow[row] = rm; Erow[row] = re; Lrow[row] = lc;
    }
    __syncthreads();                  // B3: guards red[] + buf reuse at iter i+2
  }
}

// ---- per-class margin: m[c] = 0.5*sqrt(biased var over the class block) --
__global__ void classK(const float* __restrict__ g_sum, const float* __restrict__ g_sq,
                       const float* __restrict__ g_cnt, float* __restrict__ m) {
  int c = blockIdx.x * blockDim.x + threadIdx.x;
  if (c < NCOLS) {
    float cnt  = fmaxf(g_cnt[c] * (float)NCOLS, 1.0f);
    float mean = g_sum[c] / cnt;
    float var  = fmaxf(g_sq[c] / cnt - mean * mean, 0.0f);
    m[c] = 0.5f * sqrtf(var);
  }
}

// ---- O(N) epilogue: swap target term under the original-max LSE shift ----
// val = (x_t - m)/S (x_t>0) or (x_t - m)*S; since m>=0, S>1: val <= x_t <= M,
// so the original row max M remains a numerically safe LSE shift.
__global__ __launch_bounds__(1024) void finalK(
    const int* __restrict__ lbl, const float* __restrict__ m,
    const float* __restrict__ Mrow, const float* __restrict__ Erow,
    const float* __restrict__ Lrow, float* __restrict__ out) {
  __shared__ float red[32];
  const int t = threadIdx.x;
  float acc = 0.f;
  for (int r = t; r < NROWS; r += 1024) {
    int cl = lbl[r];
    float mm = m[cl];
    float lc = Lrow[r];
    float val = (lc > 0.f) ? (lc - mm) / S_CONST : (lc - mm) * S_CONST;
    float M = Mrow[r], E = Erow[r];
    float Ep = E - __expf(lc - M) + __expf(val - M);  // exact cancel of old target term
    acc += logf(Ep) + M - val;
  }
  acc = waveSum(acc);
  if ((t & 31) == 0) red[t >> 5] = acc;
  __syncthreads();
  if (t < 32) {
    float x = red[t];
    x = waveSum(x);
    if (t == 0) out[0] = x / (float)NROWS;
  }
}

extern "C" void kernel_launch(void* const* d_in, const int* in_sizes, int n_in,
                              void* d_out, int out_size, void* d_ws, size_t ws_size,
                              hipStream_t stream) {
  const float* X   = (const float*)d_in[0];
  const int*   lbl = (const int*)d_in[1];

  float* ws    = (float*)d_ws;            // ~426 KB total
  float* g_sum = ws;
  float* g_sq  = ws + NCOLS;
  float* g_cnt = ws + 2 * NCOLS;
  float* m     = ws + 3 * NCOLS;
  float* Mr    = ws + 4 * NCOLS;
  float* Er    = Mr + NROWS;
  float* Lr    = Er + NROWS;

  zeroK<<<(3 * NCOLS + 255) / 256, 256, 0, stream>>>(g_sum, 3 * NCOLS);
  rowpass<<<NROWS / RPB, TPB, 0, stream>>>(X, lbl, g_sum, g_sq, g_cnt, Mr, Er, Lr);
  classK<<<NCOLS / 256, 256, 0, stream>>>(g_sum, g_sq, g_cnt, m);
  finalK<<<1, 1024, 0, stream>>>(lbl, m, Mr, Er, Lr, (float*)d_out);
}